// TFLOPLoss_80101140070992
// MI455X (gfx1250) — compile-verified
//
#include <hip/hip_runtime.h>

// ---------------------------------------------------------------------------
// Problem constants (from reference)
// ---------------------------------------------------------------------------
#define BB 4
#define LL 2048
#define VV 8192
#define NN 688
#define NEG_INF_F (-1e9f)

typedef float f4  __attribute__((ext_vector_type(4)));
typedef float v2f __attribute__((ext_vector_type(2)));
typedef float v8f __attribute__((ext_vector_type(8)));

// ---------------------------------------------------------------------------
// Wave32 / block reductions (deterministic fixed trees)
// ---------------------------------------------------------------------------
__device__ __forceinline__ float wredSum(float v) {
#pragma unroll
  for (int o = 16; o; o >>= 1) v += __shfl_xor(v, o, 32);
  return v;
}
__device__ __forceinline__ float wredMax(float v) {
#pragma unroll
  for (int o = 16; o; o >>= 1) v = fmaxf(v, __shfl_xor(v, o, 32));
  return v;
}

// blockDim.x == 256 (8 wave32s). Returns result to all threads.
__device__ __forceinline__ float bredSum(float v, float* sm) {
  const int lane = threadIdx.x & 31, w = threadIdx.x >> 5;
  v = wredSum(v);
  if (lane == 0) sm[w] = v;
  __syncthreads();
  if (w == 0) {
    float x = (lane < 8) ? sm[lane] : 0.0f;
    x = wredSum(x);
    if (lane == 0) sm[0] = x;
  }
  __syncthreads();
  float r = sm[0];
  __syncthreads();
  return r;
}
__device__ __forceinline__ float bredMax(float v, float* sm) {
  const int lane = threadIdx.x & 31, w = threadIdx.x >> 5;
  v = wredMax(v);
  if (lane == 0) sm[w] = v;
  __syncthreads();
  if (w == 0) {
    float x = (lane < 8) ? sm[lane] : -3.4e38f;
    x = wredMax(x);
    if (lane == 0) sm[0] = x;
  }
  __syncthreads();
  float r = sm[0];
  __syncthreads();
  return r;
}

// ---------------------------------------------------------------------------
// Kernel 1: token classification loss  (streams 256 MiB -> HBM bound)
// one block per (b,l) row; V=8192 cols; 32 floats per thread kept in VGPRs.
// __expf keeps this at the bandwidth roofline (native v_exp_f32, args <= 0).
// ---------------------------------------------------------------------------
__global__ __launch_bounds__(256) void cls_kernel(
    const float* __restrict__ tag, const int* __restrict__ tokens,
    float* __restrict__ nll_out, float* __restrict__ cnt_out) {
  __shared__ float sm[8];
  const int row = blockIdx.x;                       // [0, B*L)
  const float* rp = tag + (size_t)row * VV;
  const int tid = threadIdx.x;

  f4 vals[8];
  float mx = -3.4e38f;
#pragma unroll
  for (int j = 0; j < 8; ++j) {
    f4 v = __builtin_nontemporal_load((const f4*)(rp + j * 1024 + tid * 4));
    vals[j] = v;
    mx = fmaxf(mx, fmaxf(fmaxf(v.x, v.y), fmaxf(v.z, v.w)));
  }
  mx = bredMax(mx, sm);

  float s = 0.0f;
#pragma unroll
  for (int j = 0; j < 8; ++j) {
    f4 v = vals[j];
    s += __expf(v.x - mx) + __expf(v.y - mx) + __expf(v.z - mx) + __expf(v.w - mx);
  }
  s = bredSum(s, sm);

  if (tid == 0) {
    const int t = tokens[row];
    const float xt = rp[t];                         // single cached gather
    const float nll = (mx + __logf(s)) - xt;        // -log_softmax[t]
    const float valid = (t >= 4) ? 1.0f : 0.0f;     // specials are {0,1,2,3}
    nll_out[row] = nll * valid;
    cnt_out[row] = valid;
  }
}

// ---------------------------------------------------------------------------
// Kernel 2: pointer loss. one block per (b,n) row; L=2048 cols
// ---------------------------------------------------------------------------
__global__ __launch_bounds__(256) void ptr_kernel(
    const float* __restrict__ pl, const int* __restrict__ box,
    const int* __restrict__ dtm, float* __restrict__ out) {
  __shared__ float sm[8];
  const int g = blockIdx.x;                         // [0, B*N)
  const int b = g / NN;
  const float* rp = pl + (size_t)g * LL;
  const int* mrow = dtm + b * LL;
  const int tid = threadIdx.x;

  float m[8];
  float mx = -3.4e38f;
#pragma unroll
  for (int j = 0; j < 2; ++j) {
    const int base = j * 1024 + tid * 4;
    f4 v = __builtin_nontemporal_load((const f4*)(rp + base));
#pragma unroll
    for (int k = 0; k < 4; ++k) {
      const float mm = (mrow[base + k] != 0) ? v[k] : NEG_INF_F;
      m[j * 4 + k] = mm;
      mx = fmaxf(mx, mm);
    }
  }
  mx = bredMax(mx, sm);

  float s = 0.0f;
#pragma unroll
  for (int i = 0; i < 8; ++i) s += __expf(m[i] - mx);
  s = bredSum(s, sm);

  if (tid == 0) {
    const int bi = box[g];
    const bool valid = (bi != -1) && (bi < LL);
    const int idx = (bi < 0) ? 0 : bi;
    const float xm = (mrow[idx] != 0) ? rp[idx] : NEG_INF_F;
    const float tlp = xm - (mx + __logf(s));        // masked log_softmax @ idx
    out[g] = valid ? -tlp : 0.0f;
  }
}

// ---------------------------------------------------------------------------
// Kernel 3: span contrastive loss.
// One wave32 per 16-row tile (N=688, 16 | 688 so tiles never cross batches).
// Row sums (pos_sum / neg_sum) computed with V_WMMA_F32_16X16X4_F32 against a
// ones B-matrix: A layout puts lane<16 on K={c,c+1}, lane>=16 on K={c+2,c+3}
// of rows 0..15, so each lane feeds masked exp() values for its own columns.
// per-row max / cp / has_neg merge across halves with one shfl_xor(16).
// ---------------------------------------------------------------------------
__global__ __launch_bounds__(256) void span_kernel(
    const float* __restrict__ sim, const float* __restrict__ coef,
    float* __restrict__ out) {
  __shared__ float lpos[8][16];
  __shared__ float lneg[8][16];
  const int wave = threadIdx.x >> 5;
  const int lane = threadIdx.x & 31;
  const int tile = blockIdx.x * 8 + wave;
  const int NT = (BB * NN) / 16;                    // 172 tiles
  const bool active = (tile < NT);                  // wave-uniform

  const int r  = lane & 15;                         // row within tile
  const int hv = lane >> 4;                         // which K-half this lane owns
  const int g  = tile * 16 + r;                     // global row (b*N + j)
  const int j  = g % NN;                            // diagonal column
  const float* srow = sim  + (size_t)g * NN;
  const float* crow = coef + (size_t)g * NN;

  float mx = NEG_INF_F;                             // matches reference fill
  float cp = 0.0f;
  int hasneg = 0;
  v8f cpos = {0.f,0.f,0.f,0.f,0.f,0.f,0.f,0.f};
  v8f cneg = {0.f,0.f,0.f,0.f,0.f,0.f,0.f,0.f};

  if (active) {
    // ---- pass 1: row max over off-diagonal (scaled), cp, has_neg ----
    for (int c = 0; c < NN; c += 4) {
      const int k0 = c + hv * 2;
      v2f s2 = *(const v2f*)(srow + k0);
      v2f c2 = *(const v2f*)(crow + k0);
#pragma unroll
      for (int q = 0; q < 2; ++q) {
        const int col = k0 + q;
        const float sv = s2[q] * 10.0f;             // sim / temperature
        const bool off = (col != j);
        const bool pos = off && (c2[q] > 0.0f);
        mx = fmaxf(mx, off ? sv : NEG_INF_F);
        cp += pos ? c2[q] : 0.0f;
        hasneg |= (off && !pos) ? 1 : 0;
      }
    }
    mx = fmaxf(mx, __shfl_xor(mx, 16, 32));
    cp += __shfl_xor(cp, 16, 32);
    hasneg |= __shfl_xor(hasneg, 16, 32);

    // ---- pass 2: masked exp row-sums via WMMA (D = A x ones + C) ----
    v2f bones = {1.0f, 1.0f};
    for (int c = 0; c < NN; c += 4) {
      const int k0 = c + hv * 2;
      v2f s2 = *(const v2f*)(srow + k0);
      v2f c2 = *(const v2f*)(crow + k0);
      v2f ap, an;
#pragma unroll
      for (int q = 0; q < 2; ++q) {
        const int col = k0 + q;
        const float sv = s2[q] * 10.0f;
        const bool off = (col != j);
        const bool pos = off && (c2[q] > 0.0f);
        const float e = __expf(sv - mx);            // sv - mx <= 0, native-safe
        ap[q] = pos ? e : 0.0f;
        an[q] = (off && !pos) ? e : 0.0f;
      }
      cpos = __builtin_amdgcn_wmma_f32_16x16x4_f32(false, ap, false, bones,
                                                   (short)0, cpos, false, false);
      cneg = __builtin_amdgcn_wmma_f32_16x16x4_f32(false, an, false, bones,
                                                   (short)0, cneg, false, false);
    }
    // C/D layout: lane 0 holds rows 0..7 in VGPRs 0..7 (N=0 column),
    //             lane 16 holds rows 8..15.
    if (lane == 0) {
#pragma unroll
      for (int v = 0; v < 8; ++v) { lpos[wave][v] = cpos[v]; lneg[wave][v] = cneg[v]; }
    } else if (lane == 16) {
#pragma unroll
      for (int v = 0; v < 8; ++v) { lpos[wave][8 + v] = cpos[v]; lneg[wave][8 + v] = cneg[v]; }
    }
  }
  __syncthreads();

  if (active && lane < 16) {
    const float ps = lpos[wave][r];
    const float ns = lneg[wave][r];
    const bool valid = (cp > 0.0f) && (hasneg != 0);
    const float safe_cp = (cp > 0.0f) ? cp : 1.0f;
    const float pos_term = __logf(ps > 0.0f ? ps : 1.0f) + __logf(safe_cp);
    const float neg_term = __logf(ns > 0.0f ? ns : 1.0f);
    float lj = -(pos_term - neg_term) / safe_cp;
    lj = fminf(fmaxf(lj, -100.0f), 100.0f);
    out[g] = valid ? lj : 0.0f;
  }
}

// ---------------------------------------------------------------------------
// Kernel 4: finalize — empty-pointer BCE + deterministic sums of partials
// (tiny; keep fully-accurate expf/log1pf here)
// ---------------------------------------------------------------------------
__global__ __launch_bounds__(256) void finalize_kernel(
    const float* __restrict__ nll, const float* __restrict__ cnt,
    const float* __restrict__ ptrp, const float* __restrict__ rowl,
    const float* __restrict__ coll, const float* __restrict__ elog,
    const int* __restrict__ emask, float* __restrict__ out) {
  __shared__ float sm[8];
  const int tid = threadIdx.x;

  float a = 0.0f, c = 0.0f;
  for (int i = tid; i < BB * LL; i += 256) { a += nll[i]; c += cnt[i]; }
  a = bredSum(a, sm); c = bredSum(c, sm);

  float p = 0.0f;
  for (int i = tid; i < BB * NN; i += 256) p += ptrp[i];
  p = bredSum(p, sm);

  float rl = 0.0f, cl = 0.0f;
  for (int i = tid; i < BB * NN; i += 256) { rl += rowl[i]; cl += coll[i]; }
  rl = bredSum(rl, sm); cl = bredSum(cl, sm);

  float s1 = 0.0f, s2 = 0.0f;
  for (int i = tid; i < BB * LL; i += 256) {
    const float x = elog[i];
    const float z = (emask[i] != 0) ? 1.0f : 0.0f;
    // logaddexp(0,x) - x*z
    const float bce = fmaxf(x, 0.0f) + log1pf(expf(-fabsf(x))) - x * z;
    s1 += bce * z; s2 += z;
  }
  s1 = bredSum(s1, sm); s2 = bredSum(s2, sm);

  if (tid == 0) {
    const float cls   = a / fmaxf(c, 1.0f);
    const float ptrl  = p / (float)(BB * NN);
    const float empt  = s1 / fmaxf(s2, 1.0f);
    const float rowc  = rl / (float)(BB * NN);
    const float colc  = cl / (float)(BB * NN);
    out[0] = cls + ptrl + empt + 0.5f * rowc + 0.5f * colc;
  }
}

// ---------------------------------------------------------------------------
// Launch
// ---------------------------------------------------------------------------
extern "C" void kernel_launch(void* const* d_in, const int* in_sizes, int n_in,
                              void* d_out, int out_size, void* d_ws, size_t ws_size,
                              hipStream_t stream) {
  (void)in_sizes; (void)n_in; (void)out_size; (void)ws_size;
  const float* tag    = (const float*)d_in[0];   // [B,L,V]
  const float* plog   = (const float*)d_in[1];   // [B,N,L]
  const float* elog   = (const float*)d_in[2];   // [B,L]
  const float* rsim   = (const float*)d_in[3];   // [B,N,N]
  const float* csim   = (const float*)d_in[4];   // [B,N,N]
  const float* rcoef  = (const float*)d_in[5];   // [B,N,N]
  const float* ccoef  = (const float*)d_in[6];   // [B,N,N]
  const int*   tokens = (const int*)d_in[7];     // [B,L]
  const int*   box    = (const int*)d_in[8];     // [B,N]
  const int*   dtm    = (const int*)d_in[9];     // [B,L] bool
  const int*   emask  = (const int*)d_in[10];    // [B,L] bool

  float* ws     = (float*)d_ws;
  float* w_nll  = ws;                    // B*L   = 8192
  float* w_cnt  = ws + BB * LL;          // 8192
  float* w_ptr  = w_cnt + BB * LL;       // B*N   = 2752
  float* w_row  = w_ptr + BB * NN;       // 2752
  float* w_col  = w_row + BB * NN;       // 2752

  cls_kernel<<<BB * LL, 256, 0, stream>>>(tag, tokens, w_nll, w_cnt);
  ptr_kernel<<<BB * NN, 256, 0, stream>>>(plog, box, dtm, w_ptr);
  const int spanBlocks = ((BB * NN) / 16 + 7) / 8;  // 172 tiles, 8 waves/block
  span_kernel<<<spanBlocks, 256, 0, stream>>>(rsim, rcoef, w_row);
  span_kernel<<<spanBlocks, 256, 0, stream>>>(csim, ccoef, w_col);
  finalize_kernel<<<1, 256, 0, stream>>>(w_nll, w_cnt, w_ptr, w_row, w_col,
                                         elog, emask, (float*)d_out);
}